// Calculate_loss_Layer_31782757990972
// MI455X (gfx1250) — compile-verified
//
#include <hip/hip_runtime.h>
#include <hip/hip_bf16.h>
#include <math.h>

// ---------------------------------------------------------------------------
// YOLOv3 loss for MI455X (gfx1250).
//  * Bandwidth-bound (~90MB touched @ 23.3TB/s): skip all bytes gated by
//    obj==0 (99.8% of cells), read only conf/box logits on the common path.
//  * Per-(batch,layer) true-box tables staged into LDS with the CDNA5 async
//    DMA path: global_load_async_to_lds_b128 + s_wait_asynccnt (ASYNCcnt).
//  * Wave-level sum reduction done on the matrix pipe with
//    v_wmma_f32_16x16x4_f32 (B=ones -> row sums), replacing the LDS tree.
//  * Fully deterministic: no float atomics; fixed-order finalize.
// ---------------------------------------------------------------------------

#define NBOX 64
#define BLK  256
#define PARTIALS_PER_BATCH 42   // 2 + 8 + 32 blocks across the three layers

typedef __attribute__((ext_vector_type(2))) float v2f;
typedef __attribute__((ext_vector_type(8))) float v8f;

__device__ __forceinline__ float sigmoidf_(float x) {
    return 1.0f / (1.0f + expf(-x));
}
// bce_logits(t, x) = max(x,0) - x*t + log1p(exp(-|x|))
__device__ __forceinline__ float bce_(float t, float x) {
    return fmaxf(x, 0.0f) - x * t + log1pf(expf(-fabsf(x)));
}

// Wave32 sum-reduction on the WMMA pipe.
// A-vgpr0 = per-lane value, A-vgpr1 = 0, B = all ones:
//   D[m,n] = sum_k A[m,k] = v[m] + v[m+16]   (independent of B layout)
// Per-lane sum of the 8 D vgprs gives the half-wave sums in lanes 0-15 /
// 16-31; one shfl_xor(16) completes the 32-lane total in every lane.
__device__ __forceinline__ float wave_sum_wmma(float v) {
    v2f a; a.x = v;    a.y = 0.0f;
    v2f bo; bo.x = 1.0f; bo.y = 1.0f;
    v8f c = {};
    c = __builtin_amdgcn_wmma_f32_16x16x4_f32(
            /*neg_a=*/false, a, /*neg_b=*/false, bo,
            /*c_mod=*/(short)0, c, /*reuse_a=*/false, /*reuse_b=*/false);
    float s = ((c[0] + c[1]) + (c[2] + c[3])) + ((c[4] + c[5]) + (c[6] + c[7]));
    s += __shfl_xor(s, 16, 32);
    return s;
}

// ---------------------------------------------------------------------------
// Kernel 1: collect positive boxes (obj > 0.5) per (batch, layer).
// top_k over {0,1} objectness followed by a masked MAX == "the set of positive
// boxes" (order irrelevant), so LDS-atomic slot allocation is fine.
// ---------------------------------------------------------------------------
__global__ void collect_boxes_kernel(const float* __restrict__ y_true,
                                     float4* __restrict__ boxes,
                                     int* __restrict__ counts) {
    const int layer = blockIdx.x;   // 0..2
    const int b     = blockIdx.y;   // 0..31
    int HW, noff;
    if (layer == 0)      { HW = 169;  noff = 0;   }
    else if (layer == 1) { HW = 676;  noff = 169; }
    else                 { HW = 2704; noff = 845; }
    const int items = HW * 3;

    __shared__ int cnt;
    if (threadIdx.x == 0) cnt = 0;
    __syncthreads();

    for (int i = threadIdx.x; i < items; i += blockDim.x) {
        const int nl = i / 3;
        const int a  = i - nl * 3;
        const float* yt =
            y_true + (((size_t)b * 3549 + noff + nl) * 3 + a) * 85;
        if (yt[0] > 0.5f) {
            int slot = atomicAdd(&cnt, 1);
            if (slot < NBOX) {
                boxes[(size_t)(b * 3 + layer) * NBOX + slot] =
                    make_float4(yt[1], yt[2], yt[3], yt[4]);
            }
        }
    }
    __syncthreads();
    if (threadIdx.x == 0) counts[b * 3 + layer] = min(cnt, NBOX);
}

// ---------------------------------------------------------------------------
// Kernel 2: per-layer loss. One block = one chunk of (cell, anchor) items of
// one (batch, layer). Boxes are async-DMA'd into LDS (CDNA5 ASYNCcnt path).
// ---------------------------------------------------------------------------
__global__ void layer_loss_kernel(const float* __restrict__ y_true,
                                  const float* __restrict__ pred,
                                  const float4* __restrict__ boxes,
                                  const int* __restrict__ counts,
                                  float* __restrict__ partials,
                                  int HW, int W, int noff, int layer,
                                  int layerbase,
                                  float aw0, float ah0,
                                  float aw1, float ah1,
                                  float aw2, float ah2) {
    const int tid = threadIdx.x;
    const int b   = blockIdx.y;

    // ---- async stage of the 64-box table (1 KB) into LDS ------------------
    __shared__ float4 sBox[NBOX];
    if (tid < NBOX) {
        const float4* src = boxes + (size_t)(b * 3 + layer) * NBOX + tid;
        // Flat LDS pointer truncated to 32 bits == LDS byte address
        unsigned lds_off = (unsigned)(size_t)(&sBox[tid]);
        asm volatile("global_load_async_to_lds_b128 %0, %1, off"
                     :: "v"(lds_off), "v"(src)
                     : "memory");
    }
    asm volatile("s_wait_asynccnt 0" ::: "memory");
    __syncthreads();

    const int cnt = counts[b * 3 + layer];   // uniform -> scalar load

    float loss = 0.0f;
    const int items = HW * 3;
    const int i = blockIdx.x * BLK + tid;
    if (i < items) {
        const int nl = i / 3;
        const int a  = i - nl * 3;
        const int gy = nl / W;
        const int gx = nl - gy * W;
        const float fw = (float)W;

        const float aw = (a == 0) ? aw0 : ((a == 1) ? aw1 : aw2);
        const float ah = (a == 0) ? ah0 : ((a == 1) ? ah1 : ah2);

        const float* yt = y_true + (((size_t)b * 3549 + noff + nl) * 3 + a) * 85;
        const float* pp = pred   + (((size_t)b * HW + nl) * 255) + a * 85;

        const float obj = yt[0];
        const float l0 = pp[0], l1 = pp[1], l2 = pp[2], l3 = pp[3], l4 = pp[4];

        // predicted box (normalized) + conf
        const float bx = (sigmoidf_(l1) + (float)gx) / fw;
        const float by = (sigmoidf_(l2) + (float)gy) / fw;
        const float bw = expf(l3) * aw / fw;
        const float bh = expf(l4) * ah / fw;
        const float conf = sigmoidf_(l0);

        // best IoU against staged true boxes (max over a set: order-free)
        float best = -INFINITY;
        const float x0 = bx - bw * 0.5f, x1 = bx + bw * 0.5f;
        const float y0 = by - bh * 0.5f, y1 = by + bh * 0.5f;
        const float a1 = bw * bh;
        for (int k = 0; k < cnt; ++k) {
            const float4 tb = sBox[k];
            const float u0 = tb.x - tb.z * 0.5f, u1 = tb.x + tb.z * 0.5f;
            const float v0 = tb.y - tb.w * 0.5f, v1 = tb.y + tb.w * 0.5f;
            float iw = fminf(x1, u1) - fmaxf(x0, u0); iw = fmaxf(iw, 0.0f);
            float ih = fminf(y1, v1) - fmaxf(y0, v0); ih = fmaxf(ih, 0.0f);
            const float inter = iw * ih;
            best = fmaxf(best, inter / (a1 + tb.z * tb.w - inter));
        }
        const float ignore = (best < 0.5f) ? 1.0f : 0.0f;

        // confidence loss (always)
        const float bc = bce_(obj, conf);
        loss += obj * bc + (1.0f - obj) * bc * ignore;

        // obj-gated terms: read the other 160 floats only for positives
        if (obj > 0.0f) {
            const float tw = yt[3], th = yt[4];
            const float scale = 2.0f - tw * th;
            const float txx = yt[1] * fw - (float)gx;
            const float txy = yt[2] * fw - (float)gy;
            loss += obj * scale * (bce_(txx, bx) + bce_(txy, by));
            const float lwx = logf(tw / aw * 416.0f);
            const float lwy = logf(th / ah * 416.0f);
            const float dx = lwx - bw, dy = lwy - bh;
            loss += obj * scale * 0.5f * (dx * dx + dy * dy);
            float cl = 0.0f;
            for (int c = 5; c < 85; ++c)
                cl += bce_(yt[c], sigmoidf_(pp[c]));
            loss += obj * cl;
        }
    }

    // ---- reduction: WMMA wave sum + tiny LDS combine ----------------------
    // (code here is convergent: EXEC is all-1s as WMMA requires)
    const float wsumv = wave_sum_wmma(loss);

    __shared__ float wsum[BLK / 32];
    const int wave = tid >> 5;
    const int lane = tid & 31;
    if (lane == 0) wsum[wave] = wsumv;
    __syncthreads();
    if (tid == 0) {
        float t = 0.0f;
        for (int j = 0; j < BLK / 32; ++j) t += wsum[j];
        partials[(size_t)b * PARTIALS_PER_BATCH + layerbase + blockIdx.x] = t;
    }
}

// ---------------------------------------------------------------------------
// Kernel 3: fixed-order finalize -> d_out[b] (bit-deterministic).
// ---------------------------------------------------------------------------
__global__ void finalize_kernel(const float* __restrict__ partials,
                                float* __restrict__ out) {
    const int b = blockIdx.x;
    if (threadIdx.x == 0) {
        float s = 0.0f;
        for (int i = 0; i < PARTIALS_PER_BATCH; ++i)
            s += partials[(size_t)b * PARTIALS_PER_BATCH + i];
        out[b] = s;
    }
}

// ---------------------------------------------------------------------------
extern "C" void kernel_launch(void* const* d_in, const int* in_sizes, int n_in,
                              void* d_out, int out_size, void* d_ws, size_t ws_size,
                              hipStream_t stream) {
    const float* y_true  = (const float*)d_in[0];
    const float* pred_13 = (const float*)d_in[1];
    const float* pred_26 = (const float*)d_in[2];
    const float* pred_52 = (const float*)d_in[3];
    float* out = (float*)d_out;

    const int B = 32;

    // workspace layout (all offsets 16B-aligned)
    float4* d_boxes    = (float4*)d_ws;                               // 32*3*64*16 = 98304 B
    int*    d_counts   = (int*)((char*)d_ws + 98304);                 // 384 B
    float*  d_partials = (float*)((char*)d_ws + 98304 + 512);         // 32*42*4 = 5376 B

    collect_boxes_kernel<<<dim3(3, B), BLK, 0, stream>>>(y_true, d_boxes, d_counts);

    // layer 0: 13x13, items=507  -> 2 blocks/batch, partial base 0
    layer_loss_kernel<<<dim3(2, B), BLK, 0, stream>>>(
        y_true, pred_13, d_boxes, d_counts, d_partials,
        169, 13, 0, 0, 0,
        116.0f, 90.0f, 156.0f, 198.0f, 373.0f, 326.0f);

    // layer 1: 26x26, items=2028 -> 8 blocks/batch, partial base 2
    layer_loss_kernel<<<dim3(8, B), BLK, 0, stream>>>(
        y_true, pred_26, d_boxes, d_counts, d_partials,
        676, 26, 169, 1, 2,
        30.0f, 61.0f, 62.0f, 45.0f, 59.0f, 119.0f);

    // layer 2: 52x52, items=8112 -> 32 blocks/batch, partial base 10
    layer_loss_kernel<<<dim3(32, B), BLK, 0, stream>>>(
        y_true, pred_52, d_boxes, d_counts, d_partials,
        2704, 52, 845, 2, 10,
        10.0f, 13.0f, 16.0f, 30.0f, 33.0f, 23.0f);

    finalize_kernel<<<dim3(B), 32, 0, stream>>>(d_partials, out);
}